// CTCLoss_8555574854090
// MI455X (gfx1250) — compile-verified
//
#include <hip/hip_runtime.h>
#include <stdint.h>

// Problem constants (match reference setup_inputs)
#define T_DIM 2000
#define N_DIM 64
#define C_DIM 512
#define S_DIM 200
#define L_DIM (2 * S_DIM + 1)   // 401 extended labels
#define NEGV  (-1e30f)
#define BLOCK 416               // 13 wave32s; covers L=401 positions
#define COPY_THREADS 128        // 128 lanes * 16B = 2048B = one 512-float row
#define NBUF 16                 // LDS row ring (15 steps of prefetch lookahead)
#define LOOKAHEAD 15            // rows in flight ahead of compute
// wait immediate = LOOKAHEAD-1: with <=15 outstanding per wave (in-order
// completion), "<=14 outstanding" proves the oldest (row t+1) has landed.

// CDNA5 async global->LDS copy: VDST = LDS byte offset (low 32 bits of the
// generic pointer ARE the LDS offset on gfx1250), VADDR = 64-bit global addr.
// NT hint: rows are touched exactly once; don't thrash the 192MB L2.
__device__ __forceinline__ void async_row_load(uint32_t lds_off, const float* gaddr) {
  asm volatile("global_load_async_to_lds_b128 %0, %1, off th:TH_LOAD_NT"
               :: "v"(lds_off), "v"(gaddr)
               : "memory");
}
__device__ __forceinline__ void wait_async_le14() {
  asm volatile("s_wait_asynccnt 0xE" ::: "memory");
}

__global__ __launch_bounds__(BLOCK)
void ctc_alpha_kernel(const float* __restrict__ lp,        // (T, N, C)
                      const int*   __restrict__ targets,   // (N, S)
                      const int*   __restrict__ in_len,    // (N,)
                      const int*   __restrict__ tgt_len,   // (N,)
                      float*       __restrict__ loss_out)  // (N,) loss_n / tl_n
{
  __shared__ float rowbuf[NBUF][C_DIM];        // 32 KB ring of log-prob rows
  __shared__ float alphabuf[2][L_DIM + 2];     // double-buffered alpha, 2-pad front

  const int n   = blockIdx.x;
  const int tid = threadIdx.x;
  const int s   = tid;                         // extended-label position

  const float* __restrict__ base = lp + (size_t)n * C_DIM;   // row t at base + t*N*C
  const size_t trow = (size_t)N_DIM * C_DIM;

  // Per-thread extended label + skip-transition permission (computed once)
  int  e = 0;                                  // class index ext[s] (0 = blank)
  bool allow = false;                          // skip s-2 -> s allowed
  if (s < L_DIM && (s & 1)) {
    e = targets[n * S_DIM + ((s - 1) >> 1)];
    allow = (s == 1) || (e != targets[n * S_DIM + ((s - 3) >> 1)]);
  }
  const int inlen = in_len[n];

  // ---- Prologue: issue rows 0..14 into ring slots 0..14 ----
  if (tid < COPY_THREADS) {
    const int col = tid * 4;
#pragma unroll
    for (int r = 0; r < LOOKAHEAD; ++r) {
      uint32_t loff = (uint32_t)(uintptr_t)&rowbuf[r][col];
      async_row_load(loff, base + (size_t)r * trow + col);
    }
  }
  wait_async_le14();                           // row 0 landed (per issuing wave)
  if (tid < 2) { alphabuf[0][tid] = NEGV; alphabuf[1][tid] = NEGV; }
  __syncthreads();

  // ---- t = 0: only s=0 (blank) and s=1 (first label) reachable ----
  float areg = NEGV;                           // this thread's alpha[s], in-register
  if (s < L_DIM) {
    areg = (s <= 1) ? rowbuf[0][e] : NEGV;
    alphabuf[0][s + 2] = areg;
  }
  if (tid < COPY_THREADS) {                    // issue row 15 into slot 15
    uint32_t loff = (uint32_t)(uintptr_t)&rowbuf[LOOKAHEAD][tid * 4];
    async_row_load(loff, base + (size_t)LOOKAHEAD * trow + tid * 4);
  }
  wait_async_le14();                           // row 1 landed
  __syncthreads();

  // ---- Main scan: row t lives in rowbuf[t & 15] ----
  int cur = 0;
  for (int t = 1; t < T_DIM; ++t) {
    // Prefetch row t+15 (clamped; tail writes land in the slot of the
    // already-consumed row t-1: (t+15)&15 == (t-1)&15).
    if (tid < COPY_THREADS) {
      int pf = t + LOOKAHEAD; if (pf > T_DIM - 1) pf = T_DIM - 1;
      uint32_t loff = (uint32_t)(uintptr_t)&rowbuf[(t + LOOKAHEAD) & (NBUF - 1)][tid * 4];
      async_row_load(loff, base + (size_t)pf * trow + tid * 4);
    }
    const int nxt = cur ^ 1;
    if (s < L_DIM) {
      float a1  = areg;                               // stay
      float a2  = alphabuf[cur][s + 1];               // from s-1
      float a3  = allow ? alphabuf[cur][s] : NEGV;    // from s-2
      float lpv = rowbuf[t & (NBUF - 1)][e];          // gathered log-prob (LDS)
      float m   = fmaxf(a1, fmaxf(a2, a3));
      float sum = __expf(a1 - m) + __expf(a2 - m) + __expf(a3 - m);
      float v   = m + __logf(sum) + lpv;
      float nv  = (t < inlen) ? v : a1;               // freeze past input length
      alphabuf[nxt][s + 2] = nv;
      areg = nv;
    }
    cur = nxt;
    wait_async_le14();                         // row t+1 landed (per issuing wave)
    __syncthreads();                           // alpha[nxt] + fresh row visible
  }

  // ---- loss_n = -logaddexp(alpha_T[2tl-1], alpha_T[2tl]) / tl ----
  if (tid == 0) {
    const int tl = tgt_len[n];
    float a1 = alphabuf[cur][(2 * tl - 1) + 2];
    float a2 = alphabuf[cur][(2 * tl)     + 2];
    float m  = fmaxf(a1, a2);
    float lae = m + __logf(__expf(a1 - m) + __expf(a2 - m));
    loss_out[n] = -lae / (float)tl;
  }
}

// Deterministic mean over the 64 per-batch losses.
__global__ void ctc_reduce_kernel(const float* __restrict__ losses,
                                  float* __restrict__ out) {
  __shared__ float sm[N_DIM];
  const int tid = threadIdx.x;
  sm[tid] = losses[tid];
  __syncthreads();
  if (tid == 0) {
    float acc = 0.0f;
#pragma unroll
    for (int i = 0; i < N_DIM; ++i) acc += sm[i];
    out[0] = acc / (float)N_DIM;
  }
}

extern "C" void kernel_launch(void* const* d_in, const int* in_sizes, int n_in,
                              void* d_out, int out_size, void* d_ws, size_t ws_size,
                              hipStream_t stream) {
  const float* lp      = (const float*)d_in[0];  // (T, N, C) fp32
  const int*   targets = (const int*)  d_in[1];  // (N, S)
  const int*   in_len  = (const int*)  d_in[2];  // (N,)
  const int*   tgt_len = (const int*)  d_in[3];  // (N,)
  float* losses = (float*)d_ws;                  // 64 floats of scratch
  float* out    = (float*)d_out;                 // scalar

  ctc_alpha_kernel<<<N_DIM, BLOCK, 0, stream>>>(lp, targets, in_len, tgt_len, losses);
  ctc_reduce_kernel<<<1, N_DIM, 0, stream>>>(losses, out);
}